// OrthogonalLayer_62843961475342
// MI455X (gfx1250) — compile-verified
//
#include <hip/hip_runtime.h>
#include <hip/hip_bf16.h>

// Blocked Gram-Schmidt (BCGS2) for complex MGS, mapped onto CDNA5 WMMA.
//  - Inter-panel projections = complex GEMMs on V_WMMA_F32_16X16X4_F32
//  - Split-K with deterministic two-phase reduction (no float atomics)
//  - Intra-panel MGS = sequential VALU kernels with LDS tree reductions
// Q (= d_out, 128 MB) stays L2-resident on MI455X (192 MB L2).

typedef __attribute__((ext_vector_type(2))) float v2f;
typedef __attribute__((ext_vector_type(8))) float v8f;

#define WMMA4(A, B, C) \
  __builtin_amdgcn_wmma_f32_16x16x4_f32(false, (A), false, (B), (short)0, (C), false, false)

constexpr int MROWS  = 65536;
constexpr int RCOLS  = 256;
constexpr int PANEL  = 16;
constexpr int NCHUNK = 64;                 // split-K chunks
constexpr int CHROWS = MROWS / NCHUNK;     // 1024 rows per chunk
constexpr int NB     = 128;                // blocks for column kernels
constexpr int CT     = 256;                // threads for column kernels
constexpr float EPSV = 1e-8f;

// ---------------------------------------------------------------------------
// Kernel 1: split-K partial of S = Q[:, :j0]^H * src_panel   (complex)
// grid.x = j0/16 (previous panels), grid.y = NCHUNK/4, block = 128 (4 waves)
// Each wave owns one K-chunk of 1024 rows and one 16x16 complex output tile.
// ---------------------------------------------------------------------------
__global__ __launch_bounds__(128)
void og_proj_partial(const float2* __restrict__ Q,
                     const float2* __restrict__ src, int srcStride, int srcColOff,
                     float2* __restrict__ Spart)
{
  const int q0    = blockIdx.x * PANEL;
  const int wave  = threadIdx.x >> 5;
  const int chunk = blockIdx.y * 4 + wave;
  const int lane  = threadIdx.x & 31;
  const int li    = lane & 15;            // A: M index / B: N index
  const int kb    = (lane >> 4) << 1;     // A/B: K sub-index {kb, kb+1}
  const int mBase = chunk * CHROWS;

  v8f cRe = {};  // Qr^T Vr + Qi^T Vi
  v8f cI1 = {};  // Qr^T Vi
  v8f cI2 = {};  // Qi^T Vr

  const float2* qp = Q   + (size_t)(mBase + kb) * RCOLS     + q0 + li;
  const float2* vp = src + (size_t)(mBase + kb) * srcStride + srcColOff + li;

  for (int m = 0; m < CHROWS; m += 4) {
    const float2 a0 = qp[0];
    const float2 a1 = qp[RCOLS];
    const float2 b0 = vp[0];
    const float2 b1 = vp[srcStride];
    qp += 4 * RCOLS;
    vp += 4 * srcStride;
    v2f Ar = {a0.x, a1.x}, Ai = {a0.y, a1.y};
    v2f Br = {b0.x, b1.x}, Bi = {b0.y, b1.y};
    cRe = WMMA4(Ar, Br, cRe);
    cRe = WMMA4(Ai, Bi, cRe);
    cI1 = WMMA4(Ar, Bi, cI1);
    cI2 = WMMA4(Ai, Br, cI2);
  }

  float2* out = Spart + (size_t)chunk * (RCOLS * PANEL);
  const int rbase = (lane >> 4) * 8;      // C layout: row = r + 8*(lane>=16)
#pragma unroll
  for (int r = 0; r < 8; ++r) {
    const int qc = q0 + rbase + r;
    float2 s;
    s.x = cRe[r];
    s.y = cI1[r] - cI2[r];
    out[qc * PANEL + li] = s;
  }
}

// ---------------------------------------------------------------------------
// Kernel 2: deterministic split-K reduction  S = sum_chunks Spart
// ---------------------------------------------------------------------------
__global__ __launch_bounds__(256)
void og_proj_reduce(const float2* __restrict__ Spart, float2* __restrict__ S, int nq)
{
  const int idx = blockIdx.x * blockDim.x + threadIdx.x;
  if (idx >= nq * PANEL) return;
  float2 acc = {0.f, 0.f};
  for (int ch = 0; ch < NCHUNK; ++ch) {
    const float2 v = Spart[(size_t)ch * (RCOLS * PANEL) + idx];
    acc.x += v.x;
    acc.y += v.y;
  }
  S[idx] = acc;
}

// ---------------------------------------------------------------------------
// Kernel 3: Vtmp = src_panel - Q[:, :j0] * S   (complex GEMM, K = j0)
// grid.x = M/128, block = 256 (8 waves), each wave owns a 16-row tile.
// With j0 == 0 this is a pure panel copy.
// ---------------------------------------------------------------------------
__global__ __launch_bounds__(256)
void og_panel_update(const float2* __restrict__ Q,
                     const float2* __restrict__ S,
                     const float2* __restrict__ src, int srcStride, int srcColOff,
                     float2* __restrict__ Vtmp, int j0)
{
  const int wave = threadIdx.x >> 5;
  const int lane = threadIdx.x & 31;
  const int li   = lane & 15;
  const int kb   = (lane >> 4) << 1;
  const int m0   = (blockIdx.x * 8 + wave) * PANEL;

  v8f p1  = {};  // Qr * Sre
  v8f p2  = {};  // Qi * Sim
  v8f pim = {};  // Qr * Sim + Qi * Sre

  const float2* qp = Q + (size_t)(m0 + li) * RCOLS + kb;
  const float2* sp = S + kb * PANEL + li;

  for (int k0 = 0; k0 < j0; k0 += 4) {
    const float2 a0 = qp[0];
    const float2 a1 = qp[1];
    const float2 b0 = sp[0];
    const float2 b1 = sp[PANEL];
    qp += 4;
    sp += 4 * PANEL;
    v2f Ar = {a0.x, a1.x}, Ai = {a0.y, a1.y};
    v2f Br = {b0.x, b1.x}, Bi = {b0.y, b1.y};
    p1  = WMMA4(Ar, Br, p1);
    p2  = WMMA4(Ai, Bi, p2);
    pim = WMMA4(Ar, Bi, pim);
    pim = WMMA4(Ai, Br, pim);
  }

  const int rbase = (lane >> 4) * 8;
#pragma unroll
  for (int r = 0; r < 8; ++r) {
    const int m = m0 + rbase + r;
    const float2 xv = src[(size_t)m * srcStride + srcColOff + li];
    float2 o;
    o.x = xv.x - (p1[r] - p2[r]);
    o.y = xv.y - pim[r];
    Vtmp[(size_t)m * PANEL + li] = o;
  }
}

// ---------------------------------------------------------------------------
// Intra-panel MGS column kernels (sequential over c = 0..15)
// ---------------------------------------------------------------------------
__global__ __launch_bounds__(CT)
void og_col_dots(const float2* __restrict__ Q, const float2* __restrict__ Vtmp,
                 float2* __restrict__ dotsPart, int j0, int c)
{
  __shared__ float2 red[CT];
  const int tid = threadIdx.x;
  const int rowsPerBlock = MROWS / NB;
  const int mBase = blockIdx.x * rowsPerBlock;

  float2 acc[PANEL];
#pragma unroll
  for (int i = 0; i < PANEL; ++i) { acc[i].x = 0.f; acc[i].y = 0.f; }

  for (int m = mBase + tid; m < mBase + rowsPerBlock; m += CT) {
    const float2 v = Vtmp[(size_t)m * PANEL + c];
    const float2* qrow = Q + (size_t)m * RCOLS + j0;
#pragma unroll
    for (int i = 0; i < PANEL; ++i) {
      if (i < c) {
        const float2 q = qrow[i];
        acc[i].x += q.x * v.x + q.y * v.y;   // Re <q, v> = conj(q) . v
        acc[i].y += q.x * v.y - q.y * v.x;   // Im <q, v>
      }
    }
  }

#pragma unroll
  for (int i = 0; i < PANEL; ++i) {
    if (i < c) {                              // c is uniform across the block
      red[tid] = acc[i];
      __syncthreads();
      for (int s = CT / 2; s > 0; s >>= 1) {
        if (tid < s) { red[tid].x += red[tid + s].x; red[tid].y += red[tid + s].y; }
        __syncthreads();
      }
      if (tid == 0) dotsPart[blockIdx.x * PANEL + i] = red[0];
      __syncthreads();
    }
  }
}

__global__ __launch_bounds__(CT)
void og_col_update(const float2* __restrict__ Q, float2* __restrict__ Vtmp,
                   const float2* __restrict__ dotsPart, float* __restrict__ normPart,
                   int j0, int c)
{
  __shared__ float2 dots[PANEL];
  __shared__ float red[CT];
  const int tid = threadIdx.x;

  if (tid < c) {                               // finalize dot reductions
    float2 a = {0.f, 0.f};
    for (int b = 0; b < NB; ++b) {
      const float2 v = dotsPart[b * PANEL + tid];
      a.x += v.x; a.y += v.y;
    }
    dots[tid] = a;
  }
  __syncthreads();

  const int rowsPerBlock = MROWS / NB;
  const int mBase = blockIdx.x * rowsPerBlock;
  float nacc = 0.f;

  for (int m = mBase + tid; m < mBase + rowsPerBlock; m += CT) {
    float2 v = Vtmp[(size_t)m * PANEL + c];
    const float2* qrow = Q + (size_t)m * RCOLS + j0;
#pragma unroll
    for (int i = 0; i < PANEL; ++i) {
      if (i < c) {
        const float2 q = qrow[i];
        const float2 d = dots[i];
        v.x -= d.x * q.x - d.y * q.y;          // v -= d * q (complex)
        v.y -= d.x * q.y + d.y * q.x;
      }
    }
    Vtmp[(size_t)m * PANEL + c] = v;
    nacc += v.x * v.x + v.y * v.y;
  }

  red[tid] = nacc;
  __syncthreads();
  for (int s = CT / 2; s > 0; s >>= 1) {
    if (tid < s) red[tid] += red[tid + s];
    __syncthreads();
  }
  if (tid == 0) normPart[blockIdx.x] = red[0];
}

__global__ __launch_bounds__(CT)
void og_col_scale(float2* __restrict__ Q, const float2* __restrict__ Vtmp,
                  const float* __restrict__ normPart, int j0, int c)
{
  float s = 0.f;
  for (int b = 0; b < NB; ++b) s += normPart[b];   // redundant but deterministic
  const float inv = rsqrtf(s + EPSV);              // q = v / sqrt(|v|^2 + eps)

  const int tid = threadIdx.x;
  const int rowsPerBlock = MROWS / NB;
  const int mBase = blockIdx.x * rowsPerBlock;
  for (int m = mBase + tid; m < mBase + rowsPerBlock; m += CT) {
    float2 v = Vtmp[(size_t)m * PANEL + c];
    v.x *= inv;
    v.y *= inv;
    Q[(size_t)m * RCOLS + j0 + c] = v;
  }
}

// ---------------------------------------------------------------------------
// Host orchestration
// ---------------------------------------------------------------------------
extern "C" void kernel_launch(void* const* d_in, const int* in_sizes, int n_in,
                              void* d_out, int out_size, void* d_ws, size_t ws_size,
                              hipStream_t stream) {
  (void)in_sizes; (void)n_in; (void)out_size; (void)ws_size;

  const float2* X = (const float2*)d_in[0];   // (M, R) complex as float2
  float2* Q = (float2*)d_out;                 // (M, R) complex as float2

  char* ws = (char*)d_ws;
  const size_t SPART_B = (size_t)NCHUNK * RCOLS * PANEL * sizeof(float2); // 2 MB
  const size_t SFIN_B  = (size_t)RCOLS * PANEL * sizeof(float2);          // 32 KB
  const size_t VTMP_B  = (size_t)MROWS * PANEL * sizeof(float2);          // 8 MB
  const size_t DOTS_B  = (size_t)NB * PANEL * sizeof(float2);             // 16 KB

  float2* Spart    = (float2*)(ws);
  float2* Sfin     = (float2*)(ws + SPART_B);
  float2* Vtmp     = (float2*)(ws + SPART_B + SFIN_B);
  float2* dotsPart = (float2*)(ws + SPART_B + SFIN_B + VTMP_B);
  float*  normPart = (float*)(ws + SPART_B + SFIN_B + VTMP_B + DOTS_B);

  for (int p = 0; p < RCOLS / PANEL; ++p) {
    const int j0 = p * PANEL;

    // Pass 1: project X panel against Q[:, :j0]
    if (j0 > 0) {
      og_proj_partial<<<dim3(p, NCHUNK / 4), 128, 0, stream>>>(Q, X, RCOLS, j0, Spart);
      const int n = j0 * PANEL;
      og_proj_reduce<<<(n + 255) / 256, 256, 0, stream>>>(Spart, Sfin, j0);
    }
    og_panel_update<<<MROWS / (8 * PANEL), 256, 0, stream>>>(Q, Sfin, X, RCOLS, j0, Vtmp, j0);

    // Pass 2 (BCGS2 reorthogonalization): project Vtmp again
    if (j0 > 0) {
      og_proj_partial<<<dim3(p, NCHUNK / 4), 128, 0, stream>>>(Q, Vtmp, PANEL, 0, Spart);
      const int n = j0 * PANEL;
      og_proj_reduce<<<(n + 255) / 256, 256, 0, stream>>>(Spart, Sfin, j0);
      og_panel_update<<<MROWS / (8 * PANEL), 256, 0, stream>>>(Q, Sfin, Vtmp, PANEL, 0, Vtmp, j0);
    }

    // Intra-panel MGS (sequential columns)
    for (int c = 0; c < PANEL; ++c) {
      if (c > 0)
        og_col_dots<<<NB, CT, 0, stream>>>(Q, Vtmp, dotsPart, j0, c);
      og_col_update<<<NB, CT, 0, stream>>>(Q, Vtmp, dotsPart, normPart, j0, c);
      og_col_scale<<<NB, CT, 0, stream>>>(Q, Vtmp, normPart, j0, c);
    }
  }
}